// TopKRouter_14972255994097
// MI455X (gfx1250) — compile-verified
//
#include <hip/hip_runtime.h>
#include <math.h>
#include <stdint.h>

typedef __attribute__((ext_vector_type(2))) float v2f;
typedef __attribute__((ext_vector_type(8))) float v8f;

#define D_MODEL 2048
#define NUM_EXPERTS 64
#define TOK_PER_BLOCK 128
#define KC 32                      // K-chunk in floats
#define NCHUNK (D_MODEL / KC)      // 64
#define XP_ROW 36                  // floats per staged-x row: 32 data + 4 pad (144 B, 16B-aligned)
#define WQ_ROW 36
#define XP_FLOATS (TOK_PER_BLOCK * XP_ROW)   // 4608
#define WQ_FLOATS (NUM_EXPERTS * WQ_ROW)     // 2304
#define BUF_FLOATS (XP_FLOATS + WQ_FLOATS)   // 6912 floats = 27,648 B per buffer
#define SMEM_FLOATS (2 * BUF_FLOATS)         // 13,824 floats = 55,296 B (double buffer)
#define LOGIT_ROW 65                         // padded logits row for the top-2 scan

#define USE_ASYNC_LDS 1

#if USE_ASYNC_LDS
// GLOBAL_LOAD_ASYNC_TO_LDS_B128: per-lane 16B global->LDS copy, tracked by ASYNCcnt.
__device__ __forceinline__ void async_copy16(uint32_t lds_byte_off, const float* gptr) {
    asm volatile("global_load_async_to_lds_b128 %0, %1, off"
                 :: "v"(lds_byte_off), "v"(gptr) : "memory");
}
__device__ __forceinline__ void wait_async0() {
    asm volatile("s_wait_asynccnt 0x0" ::: "memory");
}
#endif

__global__ __launch_bounds__(256)
void topk_router_wmma(const float* __restrict__ x,
                      const float* __restrict__ Wg,
                      int*   __restrict__ idx_out,
                      float* __restrict__ w_out)
{
    __shared__ float smem[SMEM_FLOATS];

    const int tid  = threadIdx.x;
    const int lane = tid & 31;
    const int wave = tid >> 5;        // 0..7, each wave owns 16 tokens
    const int half = lane >> 4;       // K-pair select within a WMMA k-step
    const int r16  = lane & 15;       // M (token) / N (expert) index within tile
    const long tokBlock = (long)blockIdx.x * TOK_PER_BLOCK;

    // Accumulators: 4 expert tiles of 16 (C/D = 16x16 f32, 8 VGPRs each)
    v8f acc0 = {}, acc1 = {}, acc2 = {}, acc3 = {};

    // Staging assignments (256 threads; 6 x 16B copies per thread per chunk)
    const int   xrow  = tid >> 1;                 // 0..127 token row
    const int   xhalf = tid & 1;                  // which 16-float half of the 32-float chunk
    const float* xg = x + (tokBlock + xrow) * D_MODEL + xhalf * 16;
    const int   we = tid >> 2;                    // 0..63 expert row
    const int   wj = tid & 3;                     // 8 floats each
    const float* wg = Wg + (long)we * D_MODEL + wj * 8;

    // Per-thread LDS destination byte offsets within a buffer
    const uint32_t smem_base = (uint32_t)(uintptr_t)smem;
    const uint32_t xdst = smem_base + (uint32_t)(xrow * XP_ROW + xhalf * 16) * 4u;
    const uint32_t wdst = smem_base + (uint32_t)(XP_FLOATS + we * WQ_ROW + wj * 8) * 4u;

    // ---- stage one chunk into buffer `bufsel` (async, no VGPR round-trip) ----
    auto stage = [&](int bufsel, int kb) {
        const uint32_t boff = (uint32_t)(bufsel * BUF_FLOATS) * 4u;
#if USE_ASYNC_LDS
        const float* gx = xg + kb;
        #pragma unroll
        for (int c = 0; c < 4; ++c)
            async_copy16(xdst + boff + c * 16u, gx + c * 4);
        const float* gw = wg + kb;
        #pragma unroll
        for (int c = 0; c < 2; ++c)
            async_copy16(wdst + boff + c * 16u, gw + c * 4);
#else
        float4 xv[4], wv[2];
        #pragma unroll
        for (int c = 0; c < 4; ++c) xv[c] = ((const float4*)(xg + kb))[c];
        #pragma unroll
        for (int c = 0; c < 2; ++c) wv[c] = ((const float4*)(wg + kb))[c];
        float* xd = smem + bufsel * BUF_FLOATS + xrow * XP_ROW + xhalf * 16;
        float* wd = smem + bufsel * BUF_FLOATS + XP_FLOATS + we * WQ_ROW + wj * 8;
        #pragma unroll
        for (int c = 0; c < 4; ++c) ((float4*)xd)[c] = xv[c];
        #pragma unroll
        for (int c = 0; c < 2; ++c) ((float4*)wd)[c] = wv[c];
#endif
    };

    // Prologue: fill buffer 0
    stage(0, 0);
#if USE_ASYNC_LDS
    wait_async0();
#endif
    __syncthreads();

    for (int c = 0; c < NCHUNK; ++c) {
        const float* cur = smem + (c & 1) * BUF_FLOATS;
        // Kick off next chunk's copy into the other buffer (overlaps WMMA below)
        if (c + 1 < NCHUNK) stage((c + 1) & 1, (c + 1) * KC);

        // ---- 8 k-steps of V_WMMA_F32_16X16X4_F32, 4 expert tiles each ----
        const float* xprow = cur + (wave * 16 + r16) * XP_ROW;
        const float* wq    = cur + XP_FLOATS;
        #pragma unroll
        for (int s = 0; s < 8; ++s) {
            const int kk = 2 * s + half;  // lanes 0-15 -> K={4s,4s+1}, lanes 16-31 -> K={4s+2,4s+3}
            v2f a  = *(const v2f*)(xprow + kk * 2);
            v2f b0 = *(const v2f*)(wq + ( 0 + r16) * WQ_ROW + kk * 2);
            v2f b1 = *(const v2f*)(wq + (16 + r16) * WQ_ROW + kk * 2);
            v2f b2 = *(const v2f*)(wq + (32 + r16) * WQ_ROW + kk * 2);
            v2f b3 = *(const v2f*)(wq + (48 + r16) * WQ_ROW + kk * 2);
            acc0 = __builtin_amdgcn_wmma_f32_16x16x4_f32(false, a, false, b0, (short)0, acc0, false, false);
            acc1 = __builtin_amdgcn_wmma_f32_16x16x4_f32(false, a, false, b1, (short)0, acc1, false, false);
            acc2 = __builtin_amdgcn_wmma_f32_16x16x4_f32(false, a, false, b2, (short)0, acc2, false, false);
            acc3 = __builtin_amdgcn_wmma_f32_16x16x4_f32(false, a, false, b3, (short)0, acc3, false, false);
        }

#if USE_ASYNC_LDS
        wait_async0();          // next buffer fully landed in LDS
#endif
        __syncthreads();        // all waves done reading cur / copies visible
    }

    // ---- spill logits to LDS (reusing staging space), padded rows ----
    {
        const int baseRow = wave * 16 + half * 8;  // C layout: lanes 0-15 -> M=v, lanes 16-31 -> M=v+8
        #pragma unroll
        for (int v = 0; v < 8; ++v) {
            float* row = smem + (baseRow + v) * LOGIT_ROW;
            row[ 0 + r16] = acc0[v];
            row[16 + r16] = acc1[v];
            row[32 + r16] = acc2[v];
            row[48 + r16] = acc3[v];
        }
    }
    __syncthreads();

    // ---- per-token top-2 + 2-way softmax (softmax monotonic; renorm cancels) ----
    if (tid < TOK_PER_BLOCK) {
        const float* row = smem + tid * LOGIT_ROW;
        float m1 = -3.0e38f, m2 = -3.0e38f;
        int   i1 = 0, i2 = 0;
        #pragma unroll 8
        for (int e = 0; e < NUM_EXPERTS; ++e) {
            float v = row[e];
            if (v > m1)      { m2 = m1; i2 = i1; m1 = v; i1 = e; }
            else if (v > m2) { m2 = v;  i2 = e; }
        }
        float t  = __expf(m2 - m1);      // <= 1, numerically safe
        float w1 = 1.0f / (1.0f + t);
        float w2 = 1.0f - w1;
        long g = tokBlock + tid;
        idx_out[2 * g + 0] = i1;
        idx_out[2 * g + 1] = i2;
        w_out [2 * g + 0] = w1;
        w_out [2 * g + 1] = w2;
    }
}

extern "C" void kernel_launch(void* const* d_in, const int* in_sizes, int n_in,
                              void* d_out, int out_size, void* d_ws, size_t ws_size,
                              hipStream_t stream) {
    (void)in_sizes; (void)n_in; (void)out_size; (void)d_ws; (void)ws_size;
    const float* x  = (const float*)d_in[0];   // [4,4096,2048] f32
    const float* Wg = (const float*)d_in[1];   // [64,2048] f32
    const int T = 4 * 4096;
    // Tuple output, flat-concatenated in return order:
    //   [0, 2T)  expert_indices (int32), [2T, 4T) expert_weights (f32)
    int*   idx_out = (int*)d_out;
    float* w_out   = (float*)d_out + 2 * T;
    dim3 grid(T / TOK_PER_BLOCK), block(256);
    hipLaunchKernelGGL(topk_router_wmma, grid, block, 0, stream, x, Wg, idx_out, w_out);
}